// AdaptiveTripletLoss_20426864460165
// MI455X (gfx1250) — compile-verified
//
#include <hip/hip_runtime.h>
#include <hip/hip_bf16.h>
#include <math.h>

typedef __attribute__((ext_vector_type(2))) float v2f;
typedef __attribute__((ext_vector_type(8))) float v8f;

#define MAX_C 128
#define MAX_D 512
#define FILLV (-1000000.0f)
#define EPSV  (1e-12f)

__device__ __forceinline__ float wave_sum(float v) {
  #pragma unroll
  for (int m = 16; m >= 1; m >>= 1) v += __shfl_xor(v, m, 32);
  return v;
}
__device__ __forceinline__ float wave_min(float v) {
  #pragma unroll
  for (int m = 16; m >= 1; m >>= 1) v = fminf(v, __shfl_xor(v, m, 32));
  return v;
}

// ---------------- 1) row L2 normalize + a2 = |e|^2 (wave per row, b128 loads) ----
__global__ void k_normalize(const float* __restrict__ X, float* __restrict__ E,
                            float* __restrict__ a2, int n, int d) {
  int wave = threadIdx.x >> 5, lane = threadIdx.x & 31;
  int row = blockIdx.x * (blockDim.x >> 5) + wave;
  if (row >= n) return;
  const float4* x4 = (const float4*)(X + (size_t)row * d);
  int d4 = d >> 2;
  float s = 0.f;
  for (int k = lane; k < d4; k += 32) {
    float4 v = x4[k];
    s += v.x * v.x + v.y * v.y + v.z * v.z + v.w * v.w;
  }
  s = wave_sum(s);
  float inv = 1.0f / fmaxf(sqrtf(s), EPSV);
  float4* e4 = (float4*)(E + (size_t)row * d);
  for (int k = lane; k < d4; k += 32) {
    float4 v = x4[k];
    v.x *= inv; v.y *= inv; v.z *= inv; v.w *= inv;
    e4[k] = v;
  }
  if (lane == 0) a2[row] = s * inv * inv;
}

// ---------------- 2) class centers + b2c = |center|^2 (block per class) ----------
__global__ void k_centers(const float* __restrict__ E, const int* __restrict__ T,
                          float* __restrict__ Cc, float* __restrict__ b2c,
                          int n, int d) {
  int c = blockIdx.x;      // 0..MAX_C-1 ; classes >= C get zero centers naturally
  int j = threadIdx.x;     // dim index, blockDim.x == d (<=512)
  float s = 0.f; int cnt = 0;
  for (int i = 0; i < n; ++i) {
    if (T[i] == c) { ++cnt; s += E[(size_t)i * d + j]; }
  }
  float ctr = s / fmaxf((float)cnt, 1e-6f);
  Cc[(size_t)c * d + j] = ctr;
  __shared__ float red[MAX_D];
  red[j] = ctr * ctr;
  __syncthreads();
  for (int st = blockDim.x >> 1; st > 0; st >>= 1) {
    if (j < st) red[j] += red[j + st];
    __syncthreads();
  }
  if (j == 0) b2c[c] = red[0];
}

// ---------------- 3) ordered per-class member lists (single block) ---------------
__global__ void k_members(const int* __restrict__ T, int* __restrict__ cnt,
                          int* __restrict__ off, int* __restrict__ mem, int n) {
  __shared__ int scnt[MAX_C];
  __shared__ int soff[MAX_C];
  int c = threadIdx.x;
  int k = 0;
  for (int i = 0; i < n; ++i) if (T[i] == c) ++k;
  scnt[c] = k;
  __syncthreads();
  if (c == 0) { int acc = 0; for (int q = 0; q < MAX_C; ++q) { soff[q] = acc; acc += scnt[q]; } }
  __syncthreads();
  cnt[c] = scnt[c]; off[c] = soff[c];
  int w = soff[c];
  for (int i = 0; i < n; ++i) if (T[i] == c) mem[w++] = i;   // ascending index order
}

// ------- 4) top-3 hardest positives + pos-center distance (wave per row) ---------
__global__ void k_top3(const float* __restrict__ E, const float* __restrict__ a2,
                       const int* __restrict__ T, const int* __restrict__ cnt,
                       const int* __restrict__ off, const int* __restrict__ mem,
                       float* __restrict__ dpos, float* __restrict__ validf,
                       int n, int d) {
  int wave = threadIdx.x >> 5, lane = threadIdx.x & 31;
  int i = blockIdx.x * (blockDim.x >> 5) + wave;
  if (i >= n) return;
  int c = T[i];
  int base = off[c], m = cnt[c];
  float a2i = a2[i];
  int nq = d >> 7;                    // float4s per lane (d multiple of 128; d<=512)
  float4 ri[4];
  const float4* Ei4 = (const float4*)(E + (size_t)i * d);
  for (int u = 0; u < nq; ++u) ri[u] = Ei4[u * 32 + lane];

  float v0 = FILLV, v1 = FILLV, v2 = FILLV;
  int   i0 = 0, i1 = 0, i2 = 0, np = 0;
  for (int mm = 0; mm < m; ++mm) {
    int j = mem[base + mm];
    if (mm + 1 < m) {                 // hint next member row toward the caches
      int jn = mem[base + mm + 1];
      __builtin_prefetch(E + (size_t)jn * d + lane * 4, 0, 1);
    }
    if (j == i) continue;
    const float4* Ej4 = (const float4*)(E + (size_t)j * d);
    float dot = 0.f;
    for (int u = 0; u < nq; ++u) {
      float4 e = Ej4[u * 32 + lane];
      dot += ri[u].x * e.x + ri[u].y * e.y + ri[u].z * e.z + ri[u].w * e.w;
    }
    dot = wave_sum(dot);
    float sq = a2i + a2[j] - 2.0f * dot;
    float dist = sqrtf(fmaxf(sq, EPSV));
    if (dist > v0)      { v2 = v1; i2 = i1; v1 = v0; i1 = i0; v0 = dist; i0 = j; }
    else if (dist > v1) { v2 = v1; i2 = i1; v1 = dist; i1 = j; }
    else if (dist > v2) { v2 = dist; i2 = j; }
    ++np;
  }
  if (np < 3) {                       // replicate top_k over FILL entries (lowest index first)
    int j = 0;
    for (int s = np; s < 3; ++s) {
      while (j < n && !(j == i || T[j] != c)) ++j;
      int jj = (j < n) ? j : 0;
      if (s == 0) i0 = jj; else if (s == 1) i1 = jj; else i2 = jj;
      ++j;
    }
  }
  const float4* E0 = (const float4*)(E + (size_t)i0 * d);
  const float4* E1 = (const float4*)(E + (size_t)i1 * d);
  const float4* E2 = (const float4*)(E + (size_t)i2 * d);
  float acc = 0.f;
  const float third = 1.0f / 3.0f;
  for (int u = 0; u < nq; ++u) {
    float4 e0 = E0[u * 32 + lane], e1 = E1[u * 32 + lane], e2 = E2[u * 32 + lane];
    float dx = ri[u].x - (e0.x + e1.x + e2.x) * third;
    float dy = ri[u].y - (e0.y + e1.y + e2.y) * third;
    float dz = ri[u].z - (e0.z + e1.z + e2.z) * third;
    float dw = ri[u].w - (e0.w + e1.w + e2.w) * third;
    acc += dx * dx + dy * dy + dz * dz + dw * dw;
  }
  acc = wave_sum(acc);
  if (lane == 0) { dpos[i] = sqrtf(fmaxf(acc, EPSV)); validf[i] = (np > 0) ? 1.0f : 0.0f; }
}

// ------- 5) d_all = cdist(emb, centers) via FP32 WMMA, A staged in LDS ----------
// Block = 8 waves: shared 16-row A panel in LDS (loaded once), wave w owns N-tile w.
__global__ void k_dall_wmma(const float* __restrict__ E, const float* __restrict__ Cc,
                            const float* __restrict__ a2, const float* __restrict__ b2c,
                            float* __restrict__ dall, int d) {
  __shared__ float As[16 * MAX_D];                       // 32 KB (d<=512)
  int tid = threadIdx.x;
  int Mbase = blockIdx.x * 16;
  const float4* src4 = (const float4*)(E + (size_t)Mbase * d);
  float4* dst4 = (float4*)As;
  int total4 = (16 * d) >> 2;
  for (int t = tid; t < total4; t += blockDim.x) dst4[t] = src4[t];  // b128 global -> b128 ds
  __syncthreads();

  int wave = tid >> 5, lane = tid & 31;
  int Nbase = wave * 16;                                 // 8 waves cover MAX_C=128 cols
  int half = lane >> 4, l16 = lane & 15;
  const float* Arow = As + l16 * d;                      // A: 16x4, lanes 0-15 K=0/1, 16-31 K=2/3
  const float* Bcol = Cc + (size_t)(Nbase + l16) * d;    // B: 4x16, rows striped across lanes
  int kofs = half * 2;
  v8f acc = {};
  for (int k0 = 0; k0 < d; k0 += 4) {
    v2f a, b;
    a.x = Arow[k0 + kofs];     a.y = Arow[k0 + kofs + 1];
    b.x = Bcol[k0 + kofs];     b.y = Bcol[k0 + kofs + 1];
    acc = __builtin_amdgcn_wmma_f32_16x16x4_f32(false, a, false, b, (short)0, acc,
                                                false, false);
  }
  int col = Nbase + l16;
  float b2 = b2c[col];
  #pragma unroll
  for (int r = 0; r < 8; ++r) {
    int row = Mbase + r + half * 8;                      // C/D layout: VGPR r -> M=r / M=r+8
    float sq = a2[row] + b2 - 2.0f * acc[r];
    dall[(size_t)row * MAX_C + col] = sqrtf(fmaxf(sq, EPSV));
  }
}

// ------- 6) masked per-row min over classes + softplus contribution -------------
__global__ void k_loss(const float* __restrict__ dall, const int* __restrict__ T,
                       const float* __restrict__ dpos, const float* __restrict__ validf,
                       const int* __restrict__ Cptr, float* __restrict__ contrib, int n) {
  int wave = threadIdx.x >> 5, lane = threadIdx.x & 31;
  int i = blockIdx.x * (blockDim.x >> 5) + wave;
  if (i >= n) return;
  int C = *Cptr;
  int ti = T[i];
  const float4* row4 = (const float4*)(dall + (size_t)i * MAX_C);
  float4 dv = row4[lane];                                // lane covers cols 4*lane..4*lane+3
  int c0 = lane * 4;
  float v = INFINITY;
  if (c0 + 0 < C && c0 + 0 != ti) v = fminf(v, dv.x);
  if (c0 + 1 < C && c0 + 1 != ti) v = fminf(v, dv.y);
  if (c0 + 2 < C && c0 + 2 != ti) v = fminf(v, dv.z);
  if (c0 + 3 < C && c0 + 3 != ti) v = fminf(v, dv.w);
  v = wave_min(v);
  if (lane == 0) {
    float x = dpos[i] - v;                               // POS_W=NEG_W=1
    float sp = fmaxf(x, 0.f) + log1pf(expf(-fabsf(x)));  // stable softplus
    contrib[i] = sp * validf[i];
  }
}

// ------- 7) deterministic masked mean ------------------------------------------
__global__ void k_final(const float* __restrict__ contrib, const float* __restrict__ validf,
                        float* __restrict__ out, int n) {
  __shared__ float s1[256], s2[256];
  int t = threadIdx.x;
  float a = 0.f, b = 0.f;
  for (int i = t; i < n; i += 256) { a += contrib[i]; b += validf[i]; }
  s1[t] = a; s2[t] = b;
  __syncthreads();
  for (int st = 128; st > 0; st >>= 1) {
    if (t < st) { s1[t] += s1[t + st]; s2[t] += s2[t + st]; }
    __syncthreads();
  }
  if (t == 0) out[0] = s1[0] / fmaxf(s2[0], 1.0f);
}

extern "C" void kernel_launch(void* const* d_in, const int* in_sizes, int n_in,
                              void* d_out, int out_size, void* d_ws, size_t ws_size,
                              hipStream_t stream) {
  (void)n_in; (void)out_size; (void)ws_size;
  const float* X  = (const float*)d_in[0];   // embeddings [n, d]
  const int*   T  = (const int*)d_in[1];     // targets [n]
  const int*   Cp = (const int*)d_in[2];     // num_classes (device scalar)
  int n = in_sizes[1];
  int d = in_sizes[0] / n;                   // 512

  char* w = (char*)d_ws;
  float* E       = (float*)w; w += (size_t)n * d * sizeof(float);
  float* a2      = (float*)w; w += (size_t)n * sizeof(float);
  float* Cc      = (float*)w; w += (size_t)MAX_C * d * sizeof(float);
  float* b2c     = (float*)w; w += (size_t)MAX_C * sizeof(float);
  int*   cnt     = (int*)w;   w += (size_t)MAX_C * sizeof(int);
  int*   off     = (int*)w;   w += (size_t)MAX_C * sizeof(int);
  int*   mem     = (int*)w;   w += (size_t)n * sizeof(int);
  float* dall    = (float*)w; w += (size_t)n * MAX_C * sizeof(float);
  float* dpos    = (float*)w; w += (size_t)n * sizeof(float);
  float* validf  = (float*)w; w += (size_t)n * sizeof(float);
  float* contrib = (float*)w; w += (size_t)n * sizeof(float);

  int rowBlocks = (n + 7) / 8;               // 8 waves (wave32) per 256-thread block
  k_normalize<<<rowBlocks, 256, 0, stream>>>(X, E, a2, n, d);
  k_centers  <<<MAX_C, d, 0, stream>>>(E, T, Cc, b2c, n, d);
  k_members  <<<1, MAX_C, 0, stream>>>(T, cnt, off, mem, n);
  k_top3     <<<rowBlocks, 256, 0, stream>>>(E, a2, T, cnt, off, mem, dpos, validf, n, d);
  k_dall_wmma<<<n / 16, 256, 0, stream>>>(E, Cc, a2, b2c, dall, d);
  k_loss     <<<rowBlocks, 256, 0, stream>>>(dall, T, dpos, validf, Cp, contrib, n);
  k_final    <<<1, 256, 0, stream>>>(contrib, validf, (float*)d_out, n);
}